// MultiHeadedAttention_52381421142028
// MI455X (gfx1250) — compile-verified
//
#include <hip/hip_runtime.h>
#include <math.h>

typedef float v2f __attribute__((ext_vector_type(2)));
typedef float v8f __attribute__((ext_vector_type(8)));
typedef int   v4i __attribute__((vector_size(16)));   // matches builtin param type

#define BATCH 2
#define SEQ   2048
#define DPAD  512
#define DIN   508
#define HEADS 8
#define DH    64
#define NROW  (BATCH*SEQ)   /* 4096 */

static __device__ __forceinline__ v8f wmma_f32(v2f a, v2f b, v8f c) {
  // V_WMMA_F32_16X16X4_F32 : D(16x16 f32) = A(16x4 f32) * B(4x16 f32) + C
  return __builtin_amdgcn_wmma_f32_16x16x4_f32(false, a, false, b, (short)0, c, false, false);
}

// ---- async global->LDS copy (CDNA5 GLOBAL_LOAD_ASYNC_TO_LDS_B128, ASYNCcnt) ----
#if __has_builtin(__builtin_amdgcn_global_load_async_to_lds_b128) && \
    __has_builtin(__builtin_amdgcn_s_wait_asynccnt)
#define HAVE_ASYNC_LDS 1
typedef __attribute__((address_space(1))) v4i* glb_v4p;
typedef __attribute__((address_space(3))) v4i* lds_v4p;
#endif

static __device__ __forceinline__ void cp16_to_lds(float* lds, const float* g) {
#ifdef HAVE_ASYNC_LDS
  __builtin_amdgcn_global_load_async_to_lds_b128((glb_v4p)g, (lds_v4p)lds, 0, 0);
#else
  *(float4*)lds = *(const float4*)g;
#endif
}

static __device__ __forceinline__ void wait_async_lds() {
#ifdef HAVE_ASYNC_LDS
  __builtin_amdgcn_s_wait_asynccnt(0);
#endif
}

// ---------------- Kernel 1: zero-pad 508 -> 512 ----------------
__global__ __launch_bounds__(256) void k_pad(const float* __restrict__ inp,
                                             float* __restrict__ xpad) {
  int idx = blockIdx.x * 256 + threadIdx.x;     // over NROW*DPAD
  int row = idx >> 9;
  int c   = idx & 511;
  float v = 0.f;
  if (c >= 2 && c < 2 + DIN) v = inp[(size_t)row * DIN + (c - 2)];
  xpad[(size_t)idx] = v;
}

// ---------------- Kernel 2: Q/K/V = X @ W^T (head-interleaved store) ----------------
__global__ __launch_bounds__(256) void k_qkv(const float* __restrict__ xpad,
    const float* __restrict__ Wq, const float* __restrict__ Wk,
    const float* __restrict__ Wv,
    float* __restrict__ Qo, float* __restrict__ Ko, float* __restrict__ Vo) {
  const int LXS = 520;                   // padded stride (floats) to avoid bank conflicts
  __shared__ float lX[16 * 520];
  const float* W   = (blockIdx.z == 0) ? Wq : (blockIdx.z == 1) ? Wk : Wv;
  float*       Out = (blockIdx.z == 0) ? Qo : (blockIdx.z == 1) ? Ko : Vo;
  int tid = threadIdx.x;
  int m0  = blockIdx.x * 16;

  // async DMA of the 16x512 X tile into LDS (rows contiguous in global)
  const float* xg = xpad + (size_t)m0 * DPAD;
  for (int i = tid; i < 16 * DPAD / 4; i += 256) {
    int r = i >> 7, c4 = i & 127;
    cp16_to_lds(&lX[r * LXS + c4 * 4], xg + i * 4);
  }
  wait_async_lds();
  __syncthreads();

  int lane = tid & 31, wave = tid >> 5;
  int lm = lane & 15, kb = (lane >> 4) * 2, mh = (lane >> 4) * 8;
  int n0 = wave * 64;                    // each wave: 16 rows x 64 cols
  v8f acc[4] = {};
  for (int kk = 0; kk < DPAD; kk += 4) {
    v2f a; a.x = lX[lm * LXS + kk + kb]; a.y = lX[lm * LXS + kk + kb + 1];
#pragma unroll
    for (int t = 0; t < 4; ++t) {
      int n = n0 + t * 16 + lm;
      const float* wr = W + (size_t)n * DPAD + kk + kb;   // B[k][n] = W[n][k]
      v2f wb; wb.x = wr[0]; wb.y = wr[1];
      acc[t] = wmma_f32(a, wb, acc[t]);
    }
  }
#pragma unroll
  for (int t = 0; t < 4; ++t) {
    int n = n0 + t * 16 + lm;
    int h = n >> 6, c = n & 63;
#pragma unroll
    for (int v = 0; v < 8; ++v) {
      int m = m0 + mh + v;
      int bb = m >> 11, l = m & 2047;
      Out[(((size_t)bb * HEADS + h) * SEQ + l) * DH + c] = acc[t][v];
    }
  }
}

// ---------------- Kernel 3: scores = QK^T*scale, softmax, write attn ----------------
__global__ __launch_bounds__(256) void k_scores(const float* __restrict__ Q,
    const float* __restrict__ K, float* __restrict__ attn) {
  const int SQS = 68;                    // padded Q-tile stride
  __shared__ float sQ[16 * 68];
  __shared__ float sS[16 * 2048];        // 128 KB score strip (CDNA5 320KB LDS)
  __shared__ float red[256];
  __shared__ float rmax[16], rsum[16];
  int tid = threadIdx.x;
  int m0 = blockIdx.x * 16;
  int bh = blockIdx.y;
  const float scale = 0.125f;            // 1/sqrt(64)

  // async DMA of the 16x64 Q tile (256 float4's; stride 68*4B = 272B, 16B aligned)
  {
    const float* qg = Q + ((size_t)bh * SEQ + m0) * DH;
    int r = tid >> 4, c4 = tid & 15;
    cp16_to_lds(&sQ[r * SQS + c4 * 4], qg + tid * 4);
  }
  wait_async_lds();
  __syncthreads();

  int lane = tid & 31, wave = tid >> 5;
  int lm = lane & 15, kb = (lane >> 4) * 2, mh = (lane >> 4) * 8;
  for (int t = 0; t < 16; ++t) {
    int n0c = (wave + 8 * t) * 16;
    const float* kr = K + ((size_t)bh * SEQ + n0c + lm) * DH;  // B[d][n] = K[n][d]
    v8f acc = {};
#pragma unroll
    for (int kk = 0; kk < DH; kk += 4) {
      v2f a; a.x = sQ[lm * SQS + kk + kb]; a.y = sQ[lm * SQS + kk + kb + 1];
      v2f b; b.x = kr[kk + kb];            b.y = kr[kk + kb + 1];
      acc = wmma_f32(a, b, acc);
    }
#pragma unroll
    for (int v = 0; v < 8; ++v)
      sS[(mh + v) * 2048 + n0c + lm] = acc[v] * scale;
  }
  __syncthreads();

  // row-wise softmax over 2048 columns (16 threads per row)
  int r = tid >> 4, sub = tid & 15;
  float mx = -3.4e38f;
  for (int j = 0; j < 128; ++j) mx = fmaxf(mx, sS[r * 2048 + sub + 16 * j]);
  red[tid] = mx;
  __syncthreads();
  if (sub == 0) {
    float m2 = red[r * 16];
    for (int i = 1; i < 16; ++i) m2 = fmaxf(m2, red[r * 16 + i]);
    rmax[r] = m2;
  }
  __syncthreads();
  float rm = rmax[r];
  float sm = 0.f;
  for (int j = 0; j < 128; ++j) {
    int idx = r * 2048 + sub + 16 * j;
    float e = __expf(sS[idx] - rm);
    sS[idx] = e;
    sm += e;
  }
  red[tid] = sm;
  __syncthreads();
  if (sub == 0) {
    float s2 = 0.f;
    for (int i = 0; i < 16; ++i) s2 += red[r * 16 + i];
    rsum[r] = s2;
  }
  __syncthreads();
  for (int rr = 0; rr < 16; ++rr) {            // coalesced write
    float inv = 1.f / rsum[rr];
    size_t base = ((size_t)bh * SEQ + m0 + rr) * (size_t)SEQ;
    for (int c = tid; c < SEQ; c += 256)
      attn[base + c] = sS[rr * 2048 + c] * inv;
  }
}

// ---------------- Kernel 4: O = attn @ V  (double-buffered async V chunks) ----------------
__global__ __launch_bounds__(256) void k_attnv(const float* __restrict__ attn,
    const float* __restrict__ V, float* __restrict__ Y) {
  __shared__ float sV[2][128 * DH];      // 2 x 32 KB, double-buffered
  int tid = threadIdx.x;
  int bh = blockIdx.y;
  int b = bh >> 3, h = bh & 7;
  int lane = tid & 31, wave = tid >> 5;
  int lm = lane & 15, kb = (lane >> 4) * 2, mh = (lane >> 4) * 8;
  int m0 = blockIdx.x * 128 + wave * 16;
  const float* aRow = attn + ((size_t)bh * SEQ + m0 + lm) * (size_t)SEQ;
  const float* vB   = V + (size_t)bh * SEQ * DH;

  // preload chunk 0
  for (int i = tid; i < 128 * DH / 4; i += 256)
    cp16_to_lds(&sV[0][i * 4], vB + i * 4);
  wait_async_lds();
  __syncthreads();

  v8f acc[4] = {};
  const int NCHUNK = SEQ / 128;          // 16
  for (int c = 0; c < NCHUNK; ++c) {
    int cur = c & 1;
    int kc = c * 128;
    // kick off DMA of next chunk into the other buffer (overlaps WMMA below)
    if (c + 1 < NCHUNK) {
      const float* vg = vB + (size_t)(kc + 128) * DH;
      for (int i = tid; i < 128 * DH / 4; i += 256)
        cp16_to_lds(&sV[cur ^ 1][i * 4], vg + i * 4);
    }
    const float* sVc = sV[cur];
    for (int kk = 0; kk < 128; kk += 4) {
      v2f a; a.x = aRow[kc + kk + kb]; a.y = aRow[kc + kk + kb + 1];
#pragma unroll
      for (int t = 0; t < 4; ++t) {
        int n = t * 16 + lm;
        v2f bb; bb.x = sVc[(kk + kb) * DH + n]; bb.y = sVc[(kk + kb + 1) * DH + n];
        acc[t] = wmma_f32(a, bb, acc[t]);
      }
    }
    wait_async_lds();
    __syncthreads();
  }
#pragma unroll
  for (int t = 0; t < 4; ++t) {
    int n = t * 16 + lm;
#pragma unroll
    for (int v = 0; v < 8; ++v) {
      int m = m0 + mh + v;
      Y[((size_t)b * SEQ + m) * DPAD + h * DH + n] = acc[t][v];
    }
  }
}

// ---------------- Kernel 5: residual + LayerNorm + un-pad ----------------
__global__ __launch_bounds__(256) void k_ln(const float* __restrict__ Y,
    const float* __restrict__ xpad, const float* __restrict__ gamma,
    const float* __restrict__ beta, float* __restrict__ out) {
  __shared__ float red[256];
  int row = blockIdx.x;
  int tid = threadIdx.x;
  size_t base = (size_t)row * DPAD;
  float v0 = Y[base + tid]       + xpad[base + tid];
  float v1 = Y[base + tid + 256] + xpad[base + tid + 256];
  red[tid] = v0 + v1;
  __syncthreads();
  for (int s = 128; s > 0; s >>= 1) {
    if (tid < s) red[tid] += red[tid + s];
    __syncthreads();
  }
  float mu = red[0] * (1.f / DPAD);
  __syncthreads();
  float d0 = v0 - mu, d1 = v1 - mu;
  red[tid] = d0 * d0 + d1 * d1;
  __syncthreads();
  for (int s = 128; s > 0; s >>= 1) {
    if (tid < s) red[tid] += red[tid + s];
    __syncthreads();
  }
  float inv = rsqrtf(red[0] * (1.f / DPAD) + 1e-6f);
  float o0 = gamma[tid] * (d0 * inv) + beta[tid];
  float o1 = gamma[tid + 256] * (d1 * inv) + beta[tid + 256];
  size_t ob = (size_t)row * DIN;
  int c0 = tid, c1 = tid + 256;
  if (c0 >= 2 && c0 < 2 + DIN) out[ob + c0 - 2] = o0;
  if (c1 >= 2 && c1 < 2 + DIN) out[ob + c1 - 2] = o1;
}

extern "C" void kernel_launch(void* const* d_in, const int* in_sizes, int n_in,
                              void* d_out, int out_size, void* d_ws, size_t ws_size,
                              hipStream_t stream) {
  const float* inp   = (const float*)d_in[0];
  const float* Wq    = (const float*)d_in[1];
  const float* Wk    = (const float*)d_in[2];
  const float* Wv    = (const float*)d_in[3];
  const float* gamma = (const float*)d_in[4];
  const float* beta  = (const float*)d_in[5];

  float* ws = (float*)d_ws;
  const size_t NPAD = (size_t)NROW * DPAD;        // 2M floats = 8 MB
  float* xpad = ws;
  float* Q    = ws + NPAD;
  float* K    = ws + 2 * NPAD;
  float* V    = ws + 3 * NPAD;
  float* Y    = ws + 4 * NPAD;                    // total ws use: 40 MB

  float* outp = (float*)d_out;                    // (2,2048,508) f32
  float* attn = outp + (size_t)NROW * DIN;        // (2,8,2048,2048) f32

  k_pad   <<<(int)(NPAD / 256), 256, 0, stream>>>(inp, xpad);
  k_qkv   <<<dim3(NROW / 16, 1, 3), 256, 0, stream>>>(xpad, Wq, Wk, Wv, Q, K, V);
  k_scores<<<dim3(SEQ / 16, BATCH * HEADS), 256, 0, stream>>>(Q, K, attn);
  k_attnv <<<dim3(SEQ / 128, BATCH * HEADS), 256, 0, stream>>>(attn, V, Y);
  k_ln    <<<NROW, 256, 0, stream>>>(Y, xpad, gamma, beta, outp);
}